// OriginalGINConv_28432683499905
// MI455X (gfx1250) — compile-verified
//
#include <hip/hip_runtime.h>
#include <hip/hip_bf16.h>

// GIN conv pipeline for MI455X (gfx1250, wave32, WMMA).
// N=100000 nodes (= 6250*16), IN=64, EMB=128, E=1.6M edges.
// All GEMMs use V_WMMA_F32_16X16X4_F32 (full fp32 -> bit-compatible precision
// class with the fp32 reference). Edge gather/scatter rides the 192MB L2.

#define N_ROWS  100000
#define IN_DIM  64
#define EMB_DIM 128
#define BN_EPS  1e-5f

typedef float v2f __attribute__((ext_vector_type(2)));
typedef float v8f __attribute__((ext_vector_type(8)));

// ---------------------------------------------------------------------------
// Kernel 1: edge scatter.  agg (pre-initialized to x, which provides the
// (1+eps)*x self term with eps=0) accumulates x[src]*w into row dst.
// 8 edges per 256-thread block (one wave per edge), each lane owns 2 floats.
// ---------------------------------------------------------------------------
__global__ __launch_bounds__(256) void gin_scatter(
    const float* __restrict__ x, const int* __restrict__ edge_index,
    const float* __restrict__ edge_weight, float* __restrict__ agg, int E)
{
    int e = blockIdx.x * 8 + (threadIdx.x >> 5);
    if (e >= E) return;
    int lane = threadIdx.x & 31;
    int src = edge_index[e];        // row 0 of [2,E]
    int dst = edge_index[E + e];    // row 1 of [2,E]
    float w = edge_weight[e];
    v2f v = *(const v2f*)(x + (size_t)src * IN_DIM + 2 * lane);
    float* out = agg + (size_t)dst * IN_DIM + 2 * lane;
    unsafeAtomicAdd(out,     v.x * w);   // global_atomic_add_f32, L2-resident
    unsafeAtomicAdd(out + 1, v.y * w);
}

// ---------------------------------------------------------------------------
// Kernel 2: H = agg @ W1 + b1   (100000x64 @ 64x128, fp32 WMMA)
// Block = 8 waves; block b owns row-tile b (16 rows), wave w owns col-tile w.
// A frag (16x4): lanes 0-15 -> M=lane, K={k0,k0+1}; lanes 16-31 -> K={k0+2,k0+3}.
// B frag (4x16): VGPR0 = row k0 (+2 for upper lanes), VGPR1 = row k0+1 (+2).
// C/D (16x16):   VGPR v -> row v + 8*(lane>=16), col = lane&15.
// ---------------------------------------------------------------------------
__global__ __launch_bounds__(256) void gin_gemm1(
    const float* __restrict__ A, const float* __restrict__ W1,
    const float* __restrict__ b1, float* __restrict__ H)
{
    const int lane    = threadIdx.x & 31;
    const int l16     = lane & 15;
    const int half    = lane >> 4;
    const int colBase = (threadIdx.x >> 5) * 16;   // wave id -> col tile (8 tiles)
    const int rowBase = blockIdx.x * 16;

    const float* arow = A + (size_t)(rowBase + l16) * IN_DIM + half * 2;
    v8f acc = {};
#pragma unroll
    for (int k0 = 0; k0 < IN_DIM; k0 += 4) {
        v2f a = *(const v2f*)(arow + k0);                 // A[M][k0+2*half .. +1]
        int kb = k0 + half * 2;
        v2f b;
        b.x = W1[(size_t)kb       * EMB_DIM + colBase + l16];
        b.y = W1[(size_t)(kb + 1) * EMB_DIM + colBase + l16];
        acc = __builtin_amdgcn_wmma_f32_16x16x4_f32(
            /*neg_a=*/false, a, /*neg_b=*/false, b,
            /*c_mod=*/(short)0, acc, /*reuse_a=*/false, /*reuse_b=*/false);
    }
    const float bias = b1[colBase + l16];
#pragma unroll
    for (int v = 0; v < 8; ++v) {
        int row = rowBase + v + half * 8;
        H[(size_t)row * EMB_DIM + colBase + l16] = acc[v] + bias;
    }
}

// ---------------------------------------------------------------------------
// Kernel 3: per-column BN stats folded into affine form.
// One block per column; stats[c] = scale, stats[128+c] = shift so that
// BN+ReLU(h) == max(h*scale + shift, 0).  Biased variance (jnp.var default).
// ---------------------------------------------------------------------------
__global__ __launch_bounds__(256) void gin_bnstats(
    const float* __restrict__ H, const float* __restrict__ gamma,
    const float* __restrict__ beta, float* __restrict__ stats)
{
    __shared__ float ssum[256];
    __shared__ float ssq[256];
    const int col = blockIdx.x;
    const int tid = threadIdx.x;
    float s = 0.f, q = 0.f;
    for (int r = tid; r < N_ROWS; r += 256) {
        float v = H[(size_t)r * EMB_DIM + col];
        s += v;
        q += v * v;
    }
    ssum[tid] = s; ssq[tid] = q;
    __syncthreads();
    for (int w = 128; w > 0; w >>= 1) {
        if (tid < w) { ssum[tid] += ssum[tid + w]; ssq[tid] += ssq[tid + w]; }
        __syncthreads();
    }
    if (tid == 0) {
        const float invN = 1.0f / (float)N_ROWS;
        float mean = ssum[0] * invN;
        float var  = ssq[0] * invN - mean * mean;
        float scale = gamma[col] * rsqrtf(var + BN_EPS);
        stats[col]           = scale;
        stats[EMB_DIM + col] = beta[col] - mean * scale;
    }
}

// ---------------------------------------------------------------------------
// Kernel 4: OUT = relu(H*scale + shift) @ W2 + b2  (100000x128 @ 128x128)
// BN + ReLU fused into the WMMA A-fragment load (one FMA + max per element).
// ---------------------------------------------------------------------------
__global__ __launch_bounds__(256) void gin_gemm2(
    const float* __restrict__ H, const float* __restrict__ stats,
    const float* __restrict__ W2, const float* __restrict__ b2,
    float* __restrict__ OUT)
{
    const int lane    = threadIdx.x & 31;
    const int l16     = lane & 15;
    const int half    = lane >> 4;
    const int colBase = (threadIdx.x >> 5) * 16;
    const int rowBase = blockIdx.x * 16;

    const float* hrow  = H + (size_t)(rowBase + l16) * EMB_DIM + half * 2;
    const float* scale = stats;
    const float* shift = stats + EMB_DIM;

    v8f acc = {};
#pragma unroll 8
    for (int k0 = 0; k0 < EMB_DIM; k0 += 4) {
        int kb = k0 + half * 2;               // this lane's K base
        v2f hv = *(const v2f*)(hrow + k0);    // H[M][kb], H[M][kb+1]
        v2f a;
        a.x = fmaxf(fmaf(hv.x, scale[kb],     shift[kb]),     0.0f);
        a.y = fmaxf(fmaf(hv.y, scale[kb + 1], shift[kb + 1]), 0.0f);
        v2f b;
        b.x = W2[(size_t)kb       * EMB_DIM + colBase + l16];
        b.y = W2[(size_t)(kb + 1) * EMB_DIM + colBase + l16];
        acc = __builtin_amdgcn_wmma_f32_16x16x4_f32(
            false, a, false, b, (short)0, acc, false, false);
    }
    const float bias = b2[colBase + l16];
#pragma unroll
    for (int v = 0; v < 8; ++v) {
        int row = rowBase + v + half * 8;
        OUT[(size_t)row * EMB_DIM + colBase + l16] = acc[v] + bias;
    }
}

// ---------------------------------------------------------------------------
extern "C" void kernel_launch(void* const* d_in, const int* in_sizes, int n_in,
                              void* d_out, int out_size, void* d_ws, size_t ws_size,
                              hipStream_t stream) {
    const float* x           = (const float*)d_in[0];   // [N, 64]
    const int*   edge_index  = (const int*)  d_in[1];   // [2, E]
    // d_in[2] = edge_attr, unused by the module
    const float* edge_weight = (const float*)d_in[3];   // [E]
    const float* W1          = (const float*)d_in[4];   // [64, 128]
    const float* b1          = (const float*)d_in[5];   // [128]
    const float* gamma       = (const float*)d_in[6];   // [128]
    const float* beta        = (const float*)d_in[7];   // [128]
    const float* W2          = (const float*)d_in[8];   // [128, 128]
    const float* b2          = (const float*)d_in[9];   // [128]
    float* out = (float*)d_out;

    const int n = in_sizes[0] / IN_DIM;   // 100000
    const int e = in_sizes[3];            // 1600000

    // Workspace layout: agg[N*64] | h[N*128] | stats[256]  (~76.8 MB)
    float* agg   = (float*)d_ws;
    float* h     = agg + (size_t)n * IN_DIM;
    float* stats = h   + (size_t)n * EMB_DIM;

    // agg = x  (GIN self term, eps=0) — re-initialized every call, deterministic.
    hipMemcpyAsync(agg, x, (size_t)n * IN_DIM * sizeof(float),
                   hipMemcpyDeviceToDevice, stream);

    gin_scatter<<<(e + 7) / 8, 256, 0, stream>>>(x, edge_index, edge_weight, agg, e);
    gin_gemm1  <<<n / 16,      256, 0, stream>>>(agg, W1, b1, h);
    gin_bnstats<<<EMB_DIM,     256, 0, stream>>>(h, gamma, beta, stats);
    gin_gemm2  <<<n / 16,      256, 0, stream>>>(h, stats, W2, b2, out);
    (void)n_in; (void)out_size; (void)ws_size;
}